// PostProcessor_9259949490896
// MI455X (gfx1250) — compile-verified
//
#include <hip/hip_runtime.h>
#include <hip/hip_bf16.h>
#include <cstdint>
#include <cstddef>

// Problem sizes (fixed by the reference)
#define NOBJ 8192
#define OC   151
#define NREL 1000000
#define RC   51

// d_out layout (outputs concatenated flat, all float32):
//   rel_pair_idx_sorted (2M) | pred_rel_scores (51M) | pred_rel_labels (1M)
//   | obj_scores (8192) | obj_pred (8192) | obj_class_prob (8192*151)
#define O_PAIRS   ((size_t)0)
#define O_PRS     ((size_t)2000000)
#define O_LABELS  ((size_t)53000000)
#define O_OSCORE  ((size_t)54000000)
#define O_OPRED   ((size_t)54008192)
#define O_OPROB   ((size_t)54016384)

// CDNA5 async global->LDS path (ASYNCcnt), guarded so compile never breaks.
#if defined(__gfx1250__) && __has_builtin(__builtin_amdgcn_global_load_async_to_lds_b32) && __has_builtin(__builtin_amdgcn_s_wait_asynccnt)
#define HAVE_ASYNC 1
#else
#define HAVE_ASYNC 0
#endif

#define NEG_INF (-__builtin_inff())

__device__ __forceinline__ float wave_max(float v) {
  for (int o = 16; o; o >>= 1) v = fmaxf(v, __shfl_xor(v, o, 32));
  return v;
}
__device__ __forceinline__ float wave_sum(float v) {
  for (int o = 16; o; o >>= 1) v += __shfl_xor(v, o, 32);
  return v;
}
// (value, index) max-reduce, ties -> smallest index (matches jnp.argmax)
__device__ __forceinline__ void wave_argmax(float &v, int &i) {
  for (int o = 16; o; o >>= 1) {
    float ov = __shfl_xor(v, o, 32);
    int   oi = __shfl_xor(i, o, 32);
    if (ov > v || (ov == v && oi < i)) { v = ov; i = oi; }
  }
}

// Monotone-descending 32-bit sort key for a float (ascending radix => descending x)
__device__ __forceinline__ unsigned sortkey_desc(float x) {
  unsigned b = __float_as_uint(x);
  unsigned m = (b & 0x80000000u) ? ~b : (b | 0x80000000u); // ascending-monotone
  return ~m;
}

// -------------------- Phase 1: object softmax / scores --------------------
// One wave per row (151 cols -> up to 5 per lane). 8 waves per block.
__global__ void obj_softmax_kernel(const float* __restrict__ obj_logit,
                                   float* __restrict__ out,
                                   float* __restrict__ ws_oscore) {
  const int wave = threadIdx.x >> 5, lane = threadIdx.x & 31;
  const int row = blockIdx.x * 8 + wave;
  if (row >= NOBJ) return;
  const float* rp = obj_logit + (size_t)row * OC;

  float x[5];
#pragma unroll
  for (int k = 0; k < 5; k++) {
    int c = lane + k * 32;
    x[k] = (c < OC) ? rp[c] : NEG_INF;
  }
  float m = NEG_INF;
#pragma unroll
  for (int k = 0; k < 5; k++) m = fmaxf(m, x[k]);
  m = wave_max(m);

  float e[5], s = 0.f;
#pragma unroll
  for (int k = 0; k < 5; k++) {
    int c = lane + k * 32;
    e[k] = (c < OC) ? __expf(x[k] - m) : 0.f;
    s += e[k];
  }
  s = wave_sum(s);
  const float inv = 1.0f / s;

  float* op = out + O_OPROB + (size_t)row * OC;
#pragma unroll
  for (int k = 0; k < 5; k++) {
    int c = lane + k * 32;
    if (c < OC) op[c] = (c == OC - 1) ? 0.0f : e[k] * inv; // last class zeroed
  }

  // argmax / max over foreground classes [0,150)
  float bv = NEG_INF; int bi = OC;
#pragma unroll
  for (int k = 0; k < 5; k++) {
    int c = lane + k * 32;
    if (c < OC - 1 && x[k] > bv) { bv = x[k]; bi = c; }
  }
  wave_argmax(bv, bi);
  if (lane == 0) {
    float sc = __expf(bv - m) * inv;
    out[O_OSCORE + row] = sc;
    out[O_OPRED + row]  = (float)bi;
    ws_oscore[row] = sc;
  }
}

// Stage a 51-float row through LDS via CDNA5 async-to-LDS (falls back to VMEM loads)
__device__ __forceinline__ void load_row51(const float* __restrict__ rp,
                                           float* __restrict__ sb, int lane,
                                           float &x0, float &x1) {
#if HAVE_ASYNC
  if (lane < RC)
    __builtin_amdgcn_global_load_async_to_lds_b32((int*)(rp + lane),
                                                  (int*)(sb + lane), 0, 0);
  if (lane + 32 < RC)
    __builtin_amdgcn_global_load_async_to_lds_b32((int*)(rp + lane + 32),
                                                  (int*)(sb + lane + 32), 0, 0);
  __builtin_amdgcn_s_wait_asynccnt(0);
  x0 = (lane < RC)      ? sb[lane]      : NEG_INF;
  x1 = (lane + 32 < RC) ? sb[lane + 32] : NEG_INF;
#else
  (void)sb;
  x0 = (lane < RC)      ? rp[lane]      : NEG_INF;
  x1 = (lane + 32 < RC) ? rp[lane + 32] : NEG_INF;
#endif
}

// -------------------- Phase 2: relation scores -> sort keys --------------------
__global__ void rel_score_kernel(const float* __restrict__ rel_logit,
                                 const int* __restrict__ pairs,
                                 const float* __restrict__ ws_oscore,
                                 unsigned* __restrict__ keys,
                                 unsigned* __restrict__ vals) {
  __shared__ float stage[8 * 64];
  const int wave = threadIdx.x >> 5, lane = threadIdx.x & 31;
  const int j = blockIdx.x * 8 + wave;
  if (j >= NREL) return;

  float x0, x1;
  load_row51(rel_logit + (size_t)j * RC, &stage[wave * 64], lane, x0, x1);

  float m = wave_max(fmaxf(x0, x1));
  float e0 = (lane < RC)      ? __expf(x0 - m) : 0.f;
  float e1 = (lane + 32 < RC) ? __expf(x1 - m) : 0.f;
  float s = wave_sum(e0 + e1);

  // foreground max over classes [0,50)
  float bv = NEG_INF; int bi = RC;
  if (lane < RC - 1) { bv = x0; bi = lane; }
  if (lane + 32 < RC - 1 && x1 > bv) { bv = x1; bi = lane + 32; }
  wave_argmax(bv, bi);

  if (lane == 0) {
    float rel_score = __expf(bv - m) / s;
    int p0 = pairs[2 * (size_t)j];
    int p1 = pairs[2 * (size_t)j + 1];
    float triple = rel_score * ws_oscore[p0] * ws_oscore[p1];
    keys[j] = sortkey_desc(triple);
    vals[j] = (unsigned)j;
  }
}

// -------------------- Phase 3: stable LSD radix sort (4-bit, 8 passes) ------
#define RTILE 4096  // 256 threads x 16 items

__global__ void radix_hist(const unsigned* __restrict__ keys, int n, int shift,
                           unsigned* __restrict__ hist, int nb) {
  __shared__ unsigned h[16];
  if (threadIdx.x < 16) h[threadIdx.x] = 0;
  __syncthreads();
  unsigned local[16];
#pragma unroll
  for (int b = 0; b < 16; b++) local[b] = 0;
  const int base = blockIdx.x * RTILE;
  for (int i = threadIdx.x; i < RTILE; i += 256) {
    int g = base + i;
    if (g < n) local[(keys[g] >> shift) & 15]++;
  }
#pragma unroll
  for (int b = 0; b < 16; b++)
    if (local[b]) atomicAdd(&h[b], local[b]);
  __syncthreads();
  if (threadIdx.x < 16) hist[(size_t)threadIdx.x * nb + blockIdx.x] = h[threadIdx.x];
}

__global__ void radix_scan(unsigned* __restrict__ hist, int nb) {
  __shared__ unsigned totals[16];
  const int b = threadIdx.x;
  if (b < 16) {
    unsigned run = 0;
    for (int k = 0; k < nb; k++) {
      unsigned v = hist[(size_t)b * nb + k];
      hist[(size_t)b * nb + k] = run;
      run += v;
    }
    totals[b] = run;
  }
  __syncthreads();
  if (threadIdx.x == 0) {
    unsigned run = 0;
    for (int i = 0; i < 16; i++) { unsigned v = totals[i]; totals[i] = run; run += v; }
  }
  __syncthreads();
  if (b < 16) {
    unsigned add = totals[b];
    for (int k = 0; k < nb; k++) hist[(size_t)b * nb + k] += add;
  }
}

__global__ void radix_scatter(const unsigned* __restrict__ keys,
                              const unsigned* __restrict__ vals, int n, int shift,
                              const unsigned* __restrict__ hist,
                              unsigned* __restrict__ okeys,
                              unsigned* __restrict__ ovals, int nb) {
  __shared__ unsigned cnt[16 * 257]; // pitch 257 avoids bank conflicts
  __shared__ unsigned binbase[16];
  const int t = threadIdx.x;
  const int base = blockIdx.x * RTILE + t * 16;

  unsigned k[16], v[16];
  int d[16];
  unsigned local[16];
#pragma unroll
  for (int b = 0; b < 16; b++) local[b] = 0;
#pragma unroll
  for (int i = 0; i < 16; i++) {
    int g = base + i;
    k[i] = (g < n) ? keys[g] : 0xFFFFFFFFu;
    v[i] = (g < n) ? vals[g] : 0u;
    d[i] = (int)((k[i] >> shift) & 15);
    if (g < n) local[d[i]]++;
  }
#pragma unroll
  for (int b = 0; b < 16; b++) cnt[b * 257 + t] = local[b];
  __syncthreads();
  if (t < 16) {
    unsigned run = 0;
    for (int j = 0; j < 256; j++) {
      unsigned x = cnt[t * 257 + j];
      cnt[t * 257 + j] = run;
      run += x;
    }
    binbase[t] = hist[(size_t)t * nb + blockIdx.x];
  }
  __syncthreads();
#pragma unroll
  for (int b = 0; b < 16; b++) local[b] = 0;
#pragma unroll
  for (int i = 0; i < 16; i++) {
    int g = base + i;
    if (g < n) {
      int b = d[i];
      unsigned dest = binbase[b] + cnt[b * 257 + t] + local[b]++;
      okeys[dest] = k[i];
      ovals[dest] = v[i];
    }
  }
}

// -------------------- Phase 4: gather + re-softmax in sorted order ----------
__global__ void output_kernel(const float* __restrict__ rel_logit,
                              const int* __restrict__ pairs,
                              const unsigned* __restrict__ vals,
                              float* __restrict__ out) {
  __shared__ float stage[8 * 64];
  const int wave = threadIdx.x >> 5, lane = threadIdx.x & 31;
  const int i = blockIdx.x * 8 + wave;
  if (i >= NREL) return;

  const unsigned j = vals[i];

  // prefetch the gathered row ~48 sorted positions ahead (global_prefetch_b8)
  const unsigned jn = vals[(i + 48 < NREL) ? (i + 48) : i];
  if (lane < 2)
    __builtin_prefetch(rel_logit + (size_t)jn * RC + (size_t)lane * 32, 0, 1);

  float x0, x1;
  load_row51(rel_logit + (size_t)j * RC, &stage[wave * 64], lane, x0, x1);

  float m = wave_max(fmaxf(x0, x1));
  float e0 = (lane < RC)      ? __expf(x0 - m) : 0.f;
  float e1 = (lane + 32 < RC) ? __expf(x1 - m) : 0.f;
  float s = wave_sum(e0 + e1);
  const float inv = 1.0f / s;

  float* pr = out + O_PRS + (size_t)i * RC;
  if (lane < RC)      pr[lane]      = e0 * inv;  // full 51-wide prob row
  if (lane + 32 < RC) pr[lane + 32] = e1 * inv;

  float bv = NEG_INF; int bi = RC;
  if (lane < RC - 1) { bv = x0; bi = lane; }
  if (lane + 32 < RC - 1 && x1 > bv) { bv = x1; bi = lane + 32; }
  wave_argmax(bv, bi);

  if (lane == 0) {
    out[O_LABELS + i] = (float)bi;
    out[O_PAIRS + 2 * (size_t)i]     = (float)pairs[2 * (size_t)j];
    out[O_PAIRS + 2 * (size_t)i + 1] = (float)pairs[2 * (size_t)j + 1];
  }
}

// -------------------- launcher --------------------
extern "C" void kernel_launch(void* const* d_in, const int* in_sizes, int n_in,
                              void* d_out, int out_size, void* d_ws, size_t ws_size,
                              hipStream_t stream) {
  (void)in_sizes; (void)n_in; (void)out_size; (void)ws_size;
  const float* obj_logit = (const float*)d_in[0];
  const float* rel_logit = (const float*)d_in[1];
  const int*   pairs     = (const int*)d_in[2];
  float* out = (float*)d_out;

  char* ws = (char*)d_ws;
  float*    ws_oscore = (float*)(ws);                  //   32 KB
  unsigned* keysA     = (unsigned*)(ws + 32768);       //    4 MB
  unsigned* valsA     = (unsigned*)(ws + 4032768);     //    4 MB
  unsigned* keysB     = (unsigned*)(ws + 8032768);     //    4 MB
  unsigned* valsB     = (unsigned*)(ws + 12032768);    //    4 MB
  unsigned* hist      = (unsigned*)(ws + 16032768);    //  ~16 KB

  const int nb = (NREL + RTILE - 1) / RTILE; // 245 radix blocks

  obj_softmax_kernel<<<NOBJ / 8, 256, 0, stream>>>(obj_logit, out, ws_oscore);
  rel_score_kernel<<<NREL / 8, 256, 0, stream>>>(rel_logit, pairs, ws_oscore,
                                                 keysA, valsA);

  unsigned *ks = keysA, *vs = valsA, *kd = keysB, *vd = valsB;
  for (int p = 0; p < 8; p++) {
    const int shift = p * 4;
    radix_hist<<<nb, 256, 0, stream>>>(ks, NREL, shift, hist, nb);
    radix_scan<<<1, 256, 0, stream>>>(hist, nb);
    radix_scatter<<<nb, 256, 0, stream>>>(ks, vs, NREL, shift, hist, kd, vd, nb);
    unsigned* t;
    t = ks; ks = kd; kd = t;
    t = vs; vs = vd; vd = t;
  }
  // 8 passes (even) -> sorted result back in keysA/valsA (== ks/vs)

  output_kernel<<<NREL / 8, 256, 0, stream>>>(rel_logit, pairs, vs, out);
}